// WeightedSumWrapper_42133629174021
// MI455X (gfx1250) — compile-verified
//
#include <hip/hip_runtime.h>

typedef float f32x4 __attribute__((ext_vector_type(4)));

// Grid: (T, B). One block per (b,t) output row; 256 threads x float4 = D=1024.
// Fully coalesced b128 non-temporal streaming; masked rows (t >= len[b])
// write zeros and load nothing (halves expected input traffic).
template <int L>
__global__ __launch_bounds__(256) void mix_rows_kernel(
    const float* __restrict__ x,       // [L,B,T,D]
    const float* __restrict__ mw,      // [L]
    const float* __restrict__ gamma,   // [1]
    const int*   __restrict__ lengths, // [B]
    float*       __restrict__ out,     // [B,T,D]
    int T, int D, size_t layer_stride) // layer_stride = B*T*D
{
    const int t  = blockIdx.x;         // no div/mod: straight from hardware regs
    const int b  = blockIdx.y;
    const int d0 = threadIdx.x * 4;    // 0..1020

    float* orow = out + (((size_t)b * T + t) * (size_t)D + d0);

    const int len = lengths[b];        // uniform per block (scalar load)
    if (t >= len) {
        f32x4 z = {0.f, 0.f, 0.f, 0.f};
        __builtin_nontemporal_store(z, (f32x4*)orow);
        return;
    }

    // gamma * softmax(mw): uniform scalars, scalarized by the compiler.
    float e[L];
    float m = mw[0];
#pragma unroll
    for (int l = 1; l < L; ++l) m = fmaxf(m, mw[l]);
    float s = 0.f;
#pragma unroll
    for (int l = 0; l < L; ++l) { e[l] = __expf(mw[l] - m); s += e[l]; }
    const float g = gamma[0] / s;

    // fwd half (d < D/2) reads row t; bwd half reads mirrored row len-1-t.
    // Both contiguous in d -> coalesced; branch is wave-uniform (wave32:
    // waves 0-3 take fwd, waves 4-7 take bwd).
    const int tsrc = (d0 < (D >> 1)) ? t : (len - 1 - t);
    const float* src = x + ((size_t)b * T + tsrc) * (size_t)D + d0;

    f32x4 v[L];
#pragma unroll
    for (int l = 0; l < L; ++l)
        v[l] = __builtin_nontemporal_load((const f32x4*)(src + (size_t)l * layer_stride));

    f32x4 acc = v[0] * (e[0] * g);
#pragma unroll
    for (int l = 1; l < L; ++l)
        acc += v[l] * (e[l] * g);

    __builtin_nontemporal_store(acc, (f32x4*)orow);
}

// Fallback for unexpected L (runtime loop).
__global__ __launch_bounds__(256) void mix_rows_kernel_generic(
    const float* __restrict__ x, const float* __restrict__ mw,
    const float* __restrict__ gamma, const int* __restrict__ lengths,
    float* __restrict__ out, int Lr, int T, int D, size_t layer_stride)
{
    const int t  = blockIdx.x;
    const int b  = blockIdx.y;
    const int d0 = threadIdx.x * 4;

    float* orow = out + (((size_t)b * T + t) * (size_t)D + d0);
    const int len = lengths[b];
    if (t >= len) {
        f32x4 z = {0.f, 0.f, 0.f, 0.f};
        __builtin_nontemporal_store(z, (f32x4*)orow);
        return;
    }

    float m = mw[0];
    for (int l = 1; l < Lr; ++l) m = fmaxf(m, mw[l]);
    float s = 0.f;
    for (int l = 0; l < Lr; ++l) s += __expf(mw[l] - m);
    const float g = gamma[0] / s;

    const int tsrc = (d0 < (D >> 1)) ? t : (len - 1 - t);
    const float* src = x + ((size_t)b * T + tsrc) * (size_t)D + d0;

    f32x4 acc = {0.f, 0.f, 0.f, 0.f};
    for (int l = 0; l < Lr; ++l) {
        f32x4 v = __builtin_nontemporal_load((const f32x4*)(src + (size_t)l * layer_stride));
        acc += v * (__expf(mw[l] - m) * g);
    }
    __builtin_nontemporal_store(acc, (f32x4*)orow);
}

extern "C" void kernel_launch(void* const* d_in, const int* in_sizes, int n_in,
                              void* d_out, int out_size, void* d_ws, size_t ws_size,
                              hipStream_t stream) {
    (void)n_in; (void)d_ws; (void)ws_size;
    const float* x       = (const float*)d_in[0]; // outputs [L,B,T,D]
    const float* mw      = (const float*)d_in[1]; // mix_weights [L]
    const float* gamma   = (const float*)d_in[2]; // gamma [1]
    const int*   lengths = (const int*)d_in[3];   // lengths [B]
    float*       out     = (float*)d_out;         // [B,T,D]

    const int L = in_sizes[1];
    const int B = in_sizes[3];
    const int D = 1024;                  // per reference shapes
    const int T = out_size / (B * D);
    const size_t layer_stride = (size_t)B * (size_t)T * (size_t)D;

    dim3 grid((unsigned)T, (unsigned)B); // t fast, b slow: streaming locality
    dim3 block((unsigned)(D / 4));       // 256 threads = 8 waves (wave32)

    switch (L) {
    case 2: mix_rows_kernel<2><<<grid, block, 0, stream>>>(x, mw, gamma, lengths, out, T, D, layer_stride); break;
    case 3: mix_rows_kernel<3><<<grid, block, 0, stream>>>(x, mw, gamma, lengths, out, T, D, layer_stride); break;
    case 4: mix_rows_kernel<4><<<grid, block, 0, stream>>>(x, mw, gamma, lengths, out, T, D, layer_stride); break;
    default:
        mix_rows_kernel_generic<<<grid, block, 0, stream>>>(x, mw, gamma, lengths, out, L, T, D, layer_stride);
        break;
    }
}